// UpSample1d_19696720020011
// MI455X (gfx1250) — compile-verified
//
#include <hip/hip_runtime.h>

// CDNA5 / gfx1250: fp32 WMMA 16x16x4 polyphase upsample-by-2 (kaiser-sinc K=12, depthwise)
// y[2t]   = sum_u he[u] * x[t-3+u],  he[u] = 2*filt[11-2u]
// y[2t+1] = sum_u ho[u] * x[t-2+u],  ho[u] = 2*filt[10-2u]
// Per 256-input tile, per phase:  Y[m][n] = sum_k A[m][k] * H[k][n]
//   A[m][k] = x[tileBase - off + 16m + k]  (Toeplitz tile, staged in LDS via async b128)
//   H[k][n] = h[k-n] for 0<=k-n<=5 else 0  (banded, loop-invariant, pinned in VGPRs)
// 6 chained V_WMMA_F32_16X16X4_F32 per phase; outputs packed even|odd as NT b64 stores.

typedef float v2f __attribute__((ext_vector_type(2)));
typedef float v8f __attribute__((ext_vector_type(8)));

constexpr int T_IN   = 8192;        // time length per (b,c) row
constexpr int TILE   = 256;         // input positions per wave-tile (-> 512 outputs)
constexpr int LDSW   = 272;         // staged floats per wave: x[tileBase-4 .. tileBase+268)
constexpr int CHUNKS = LDSW / 4;    // 68 b128 chunks per tile
constexpr int WAVES  = 8;           // waves per block (256 threads, wave32)
constexpr int TPW    = 4;           // tiles per wave
constexpr int TPR    = T_IN / TILE; // 32 tiles per row

__device__ __forceinline__ float pick6(int d, const float h[6]) {
    float r = 0.0f;
    r = (d == 0) ? h[0] : r;
    r = (d == 1) ? h[1] : r;
    r = (d == 2) ? h[2] : r;
    r = (d == 3) ? h[3] : r;
    r = (d == 4) ? h[4] : r;
    r = (d == 5) ? h[5] : r;
    return r;
}

__global__ __launch_bounds__(256) void upsample2x_wmma_f32(
    const float* __restrict__ x,     // [rows, T_IN]
    const float* __restrict__ filt,  // [12]
    float* __restrict__ y)           // [rows, 2*T_IN]
{
    __shared__ __align__(16) float lds[WAVES * LDSW];

    const int tid  = threadIdx.x;
    const int lane = tid & 31;
    const int wid  = tid >> 5;
    const int n    = lane & 15;   // N column (B/C/D) == A row m for this lane
    const int half = lane >> 4;   // lane half: K sub-block select / M+8

    // ---- uniform filter taps (RATIO=2 folded in) ----
    float he[6], ho[6];
#pragma unroll
    for (int u = 0; u < 6; ++u) {
        he[u] = 2.0f * filt[11 - 2 * u];
        ho[u] = 2.0f * filt[10 - 2 * u];
    }

    // ---- banded B matrices, computed ONCE; opaque so they cannot be rematerialized ----
    // B layout (4x16 f32, wave32): lanes 0-15 hold K rows {0,1} (VGPR0->K0, VGPR1->K1),
    // lanes 16-31 hold K rows {2,3}.
    float bex[6], bey[6], box[6], boy[6];
#pragma unroll
    for (int s = 0; s < 6; ++s) {
        const int k0 = 4 * s + 2 * half;
        bex[s] = pick6(k0     - n, he);
        bey[s] = pick6(k0 + 1 - n, he);
        box[s] = pick6(k0     - n, ho);
        boy[s] = pick6(k0 + 1 - n, ho);
    }
#pragma unroll
    for (int s = 0; s < 6; ++s) {
        asm volatile("" : "+v"(bex[s]), "+v"(bey[s]), "+v"(box[s]), "+v"(boy[s]));
    }
    v2f be[6], bo[6];
#pragma unroll
    for (int s = 0; s < 6; ++s) {
        be[s].x = bex[s]; be[s].y = bey[s];
        bo[s].x = box[s]; bo[s].y = boy[s];
    }

    const int  w     = blockIdx.x * WAVES + wid;   // global wave id
    const int  tile0 = w * TPW;
    const int  row   = tile0 / TPR;                // TPW | TPR -> same row for all tiles
    const float* xrow = x + (long)row * T_IN;      // uniform -> SGPR pair (saddr)
    float*       yrow = y + (long)row * (2 * T_IN);
    float* ldsW = &lds[wid * LDSW];

#pragma unroll 1
    for (int tt = 0; tt < TPW; ++tt) {
        const int tileBase = ((tile0 + tt) % TPR) * TILE;

        // Previous tile's LDS reads must retire before async writes reuse the buffer
        // (async LDS writes are not ordered with the DS pipe).
        asm volatile("s_wait_dscnt 0x0" ::: "memory");

        // ---- async global -> LDS staging, 68 x b128 chunks, saddr + 32-bit voffset ----
        // Chunks are 16B aligned and never straddle [0, T_IN); out-of-row chunks get a
        // clamped (safe) offset and are zeroed after the wait.
#pragma unroll
        for (int ii = 0; ii < 3; ++ii) {
            const int ci = ii * 32 + lane;
            if (ci < CHUNKS) {
                const int g0 = tileBase - 4 + 4 * ci;            // float index in row
                const unsigned voff =
                    ((unsigned)g0 < (unsigned)T_IN) ? (unsigned)(4 * g0) : 0u;
                // flat->LDS aperture keeps the workgroup-relative byte offset in addr[31:0]
                const unsigned ldsOff =
                    (unsigned)(unsigned long long)(const void*)&ldsW[4 * ci];
                asm volatile("global_load_async_to_lds_b128 %0, %1, %2"
                             :: "v"(ldsOff), "v"(voff), "s"(xrow)
                             : "memory");
            }
        }

        // prefetch next tile's input while the async copy is in flight
        if (tt + 1 < TPW) {
            __builtin_prefetch(&xrow[tileBase + TILE + lane * 8], 0, 0);
        }

        asm volatile("s_wait_asynccnt 0x0" ::: "memory");

        // ---- zero-fix row edges; wave-uniform guard skips interior tiles ----
        if ((tileBase == 0) || (tileBase == T_IN - TILE)) {
#pragma unroll
            for (int ii = 0; ii < 9; ++ii) {
                const int i = ii * 32 + lane;
                if (i < LDSW) {
                    const int g = tileBase - 4 + i;
                    if ((unsigned)g >= (unsigned)T_IN) ldsW[i] = 0.0f;
                }
            }
        }

        // ---- 6 chained K=4 f32 WMMAs per phase ----
        // A layout (16x4 f32): lanes 0-15: VGPR0=K(4s+0), VGPR1=K(4s+1);
        //                      lanes 16-31: VGPR0=K(4s+2), VGPR1=K(4s+3)
        // A_even[m][k] = lds[1 + 16m + k],  A_odd[m][k] = lds[2 + 16m + k]
        v8f ce = {};
        v8f co = {};
#pragma unroll
        for (int s = 0; s < 6; ++s) {
            const int k  = 4 * s + 2 * half;
            const int ib = 16 * n + k;       // n == lane&15 == A row m
            v2f ae, ao;
            ae.x = ldsW[1 + ib];
            ae.y = ldsW[2 + ib];
            ao.x = ldsW[2 + ib];
            ao.y = ldsW[3 + ib];
            ce = __builtin_amdgcn_wmma_f32_16x16x4_f32(
                     false, ae, false, be[s], (short)0, ce, false, false);
            co = __builtin_amdgcn_wmma_f32_16x16x4_f32(
                     false, ao, false, bo[s], (short)0, co, false, false);
        }

        // ---- pack even/odd phases, coalesced NT b64 stores ----
        // C/D layout: VGPR comp, lanes 0-15 -> M=comp, lanes 16-31 -> M=comp+8; N=lane&15
        // byte offset = 8*t_out = 8*(tileBase + 128*half + n) + 128*comp
        //   -> one VGPR offset per tile, comp folds into the immediate offset (GVS form)
        const unsigned ob = (unsigned)((tileBase + 128 * half + n) * 8);
#pragma unroll
        for (int comp = 0; comp < 8; ++comp) {
            union { float f[2]; unsigned long long u; } pk;
            pk.f[0] = ce[comp];   // y[2*t_out]
            pk.f[1] = co[comp];   // y[2*t_out + 1]
            __builtin_nontemporal_store(
                pk.u, (unsigned long long*)((char*)yrow + ob + comp * 128));
        }
    }
}

extern "C" void kernel_launch(void* const* d_in, const int* in_sizes, int n_in,
                              void* d_out, int out_size, void* d_ws, size_t ws_size,
                              hipStream_t stream) {
    (void)n_in; (void)out_size; (void)d_ws; (void)ws_size;
    const float* x    = (const float*)d_in[0];
    const float* filt = (const float*)d_in[1];
    float*       y    = (float*)d_out;

    const int total  = in_sizes[0];            // 8*512*8192 = 33,554,432
    const int tiles  = total / TILE;           // 131,072
    const int blocks = tiles / (WAVES * TPW);  // 4,096
    upsample2x_wmma_f32<<<blocks, 256, 0, stream>>>(x, filt, y);
}